// GGNN_16295105921633
// MI455X (gfx1250) — compile-verified
//
#include <hip/hip_runtime.h>
#include <math.h>

#define N_NODES 50000
#define N_EDGES 800000
#define F_IN    512
#define HID     64
#define NCLS    32
#define LAYERS  3

#define PAD_FIN (F_IN + 4)   // 516: (516*L) % 64 = 4L -> conflict-free float2 LDS reads
#define PADH    (HID + 4)    // 68:  (68*L)  % 64 = 4L -> conflict-free

typedef float v2f __attribute__((ext_vector_type(2)));
typedef float v8f __attribute__((ext_vector_type(8)));

// D = A(16x4 f32) x B(4x16 f32) + C, wave32.
// A frag: lanes 0-15: row=lane, k={k0,k0+1}; lanes 16-31: row=lane-16, k={k0+2,k0+3}
// B frag: lanes 0-15: col=lane, k={k0,k0+1}; lanes 16-31: col=lane-16, k={k0+2,k0+3}
// C/D:    VGPR v -> row = v + 8*(lane>=16), col = lane&15
__device__ __forceinline__ v8f wmma_f32(v2f a, v2f b, v8f c) {
    return __builtin_amdgcn_wmma_f32_16x16x4_f32(false, a, false, b, (short)0, c,
                                                 false, false);
}

// ---------------- Kernel 1: h = x @ weight_in  [50000,512]x[512,64] ----------
__global__ __launch_bounds__(128) void k_input_gemm(const float* __restrict__ x,
                                                    const float* __restrict__ w_in,
                                                    float* __restrict__ h) {
    __shared__ float As[16 * PAD_FIN];           // 33 KB: 16x512 slab of x
    const int m0  = blockIdx.x * 16;
    const int tid = threadIdx.x;

    // coalesced stage of A slab (float4, row-aligned since 512 % 4 == 0)
    for (int i = tid * 4; i < 16 * F_IN; i += 128 * 4) {
        const int row = i / F_IN, col = i % F_IN;
        const float4 v = *(const float4*)&x[(size_t)(m0 + row) * F_IN + col];
        As[row * PAD_FIN + col + 0] = v.x;
        As[row * PAD_FIN + col + 1] = v.y;
        As[row * PAD_FIN + col + 2] = v.z;
        As[row * PAD_FIN + col + 3] = v.w;
    }
    __syncthreads();

    const int wave = tid >> 5, lane = tid & 31;
    const int half = lane >> 4, li = lane & 15;
    const int n0 = wave * 16;                    // 4 waves cover N=64

    v8f acc = {};
    for (int k0 = 0; k0 < F_IN; k0 += 4) {
        const int kk = k0 + half * 2;
        v2f a; a.x = As[li * PAD_FIN + kk]; a.y = As[li * PAD_FIN + kk + 1];
        const float* bp = &w_in[(size_t)kk * HID + n0 + li];
        v2f b; b.x = bp[0]; b.y = bp[HID];
        acc = wmma_f32(a, b, acc);
    }
#pragma unroll
    for (int v = 0; v < 8; ++v) {
        const int row = v + half * 8;
        h[(size_t)(m0 + row) * HID + n0 + li] = acc[v];
    }
}

// -------- Kernel 2: m = h @ W_l  and  agg = 0  (fused) ------------------------
__global__ __launch_bounds__(128) void k_transform(const float* __restrict__ h,
                                                   const float* __restrict__ W,
                                                   float* __restrict__ m,
                                                   float* __restrict__ agg) {
    __shared__ float As[16 * PADH];
    const int m0  = blockIdx.x * 16;
    const int tid = threadIdx.x;

    for (int i = tid * 4; i < 16 * HID; i += 128 * 4) {
        const int row = i / HID, col = i % HID;
        const float4 v = *(const float4*)&h[(size_t)(m0 + row) * HID + col];
        As[row * PADH + col + 0] = v.x;
        As[row * PADH + col + 1] = v.y;
        As[row * PADH + col + 2] = v.z;
        As[row * PADH + col + 3] = v.w;
    }
    __syncthreads();

    const int wave = tid >> 5, lane = tid & 31;
    const int half = lane >> 4, li = lane & 15;
    const int n0 = wave * 16;

    v8f acc = {};
    for (int k0 = 0; k0 < HID; k0 += 4) {
        const int kk = k0 + half * 2;
        v2f a; a.x = As[li * PADH + kk]; a.y = As[li * PADH + kk + 1];
        const float* bp = &W[kk * HID + n0 + li];
        v2f b; b.x = bp[0]; b.y = bp[HID];
        acc = wmma_f32(a, b, acc);
    }
#pragma unroll
    for (int v = 0; v < 8; ++v) {
        const int row = v + half * 8;
        const size_t off = (size_t)(m0 + row) * HID + n0 + li;
        m[off]   = acc[v];
        agg[off] = 0.0f;                         // zero accumulator for scatter
    }
}

// -------- Kernel 3: agg[dst] += m[src]  (16 lanes / edge, float4 / lane) -----
__global__ __launch_bounds__(256) void k_scatter(const int* __restrict__ ei,
                                                 const float* __restrict__ m,
                                                 float* __restrict__ agg) {
    const long long gid = (long long)blockIdx.x * 256 + threadIdx.x;
    const int e = (int)(gid >> 4);
    const int c = (int)(gid & 15) * 4;
    const int s = ei[e];                          // src row
    const int d = ei[N_EDGES + e];                // dst row
    const float4 v = *(const float4*)&m[(size_t)s * HID + c];
    float* a = &agg[(size_t)d * HID + c];
    atomicAdd(a + 0, v.x);
    atomicAdd(a + 1, v.y);
    atomicAdd(a + 2, v.z);
    atomicAdd(a + 3, v.w);
}

// -------- Kernel 4: GRUCell(agg, h) -> h  (in place) -------------------------
// gi = agg @ w_ih^T + b_ih ; gh = h @ w_hh^T + b_hh ; gates [r,z,n]
__global__ __launch_bounds__(384) void k_gru(float* __restrict__ h,
                                             const float* __restrict__ agg,
                                             const float* __restrict__ w_ih,
                                             const float* __restrict__ w_hh,
                                             const float* __restrict__ b_ih,
                                             const float* __restrict__ b_hh) {
    __shared__ float Ah[16 * PADH];               // h tile
    __shared__ float Aa[16 * PADH];               // agg tile
    __shared__ float Si[16 * 192];                // gi
    __shared__ float Sh[16 * 192];                // gh
    const int m0  = blockIdx.x * 16;
    const int tid = threadIdx.x;

    for (int i = tid; i < 16 * HID; i += 384) {
        const int row = i / HID, col = i % HID;
        Ah[row * PADH + col] = h[(size_t)(m0 + row) * HID + col];
        Aa[row * PADH + col] = agg[(size_t)(m0 + row) * HID + col];
    }
    __syncthreads();

    const int wave = tid >> 5, lane = tid & 31;
    const int half = lane >> 4, li = lane & 15;
    const int j0 = wave * 16;                     // 12 waves cover 192 gate cols

    v8f ci = {}, ch = {};
    for (int k0 = 0; k0 < HID; k0 += 4) {
        const int kk = k0 + half * 2;
        v2f aA; aA.x = Aa[li * PADH + kk]; aA.y = Aa[li * PADH + kk + 1];
        v2f aH; aH.x = Ah[li * PADH + kk]; aH.y = Ah[li * PADH + kk + 1];
        // B = W^T : B[k][j] = W[j*64 + k]  -> consecutive k in memory
        const float* bip = &w_ih[(j0 + li) * HID + kk];
        const float* bhp = &w_hh[(j0 + li) * HID + kk];
        v2f bi; bi.x = bip[0]; bi.y = bip[1];
        v2f bh; bh.x = bhp[0]; bh.y = bhp[1];
        ci = wmma_f32(aA, bi, ci);
        ch = wmma_f32(aH, bh, ch);
    }
#pragma unroll
    for (int v = 0; v < 8; ++v) {
        const int row = v + half * 8, col = j0 + li;
        Si[row * 192 + col] = ci[v] + b_ih[col];
        Sh[row * 192 + col] = ch[v] + b_hh[col];
    }
    __syncthreads();

    for (int i = tid; i < 16 * HID; i += 384) {
        const int row = i / HID, c = i % HID;
        const float ir = Si[row * 192 + c],        hr = Sh[row * 192 + c];
        const float iz = Si[row * 192 + 64 + c],   hz = Sh[row * 192 + 64 + c];
        const float in_ = Si[row * 192 + 128 + c], hn = Sh[row * 192 + 128 + c];
        const float r = 1.0f / (1.0f + expf(-(ir + hr)));
        const float z = 1.0f / (1.0f + expf(-(iz + hz)));
        const float n = tanhf(in_ + r * hn);
        const float hv = Ah[row * PADH + c];
        h[(size_t)(m0 + row) * HID + c] = (1.0f - z) * n + z * hv;
    }
}

// -------- Kernel 5: out = h @ weight_out  [50000,64]x[64,32] -----------------
__global__ __launch_bounds__(64) void k_output(const float* __restrict__ h,
                                               const float* __restrict__ w_out,
                                               float* __restrict__ out) {
    __shared__ float As[16 * PADH];
    const int m0  = blockIdx.x * 16;
    const int tid = threadIdx.x;

    for (int i = tid * 4; i < 16 * HID; i += 64 * 4) {
        const int row = i / HID, col = i % HID;
        const float4 v = *(const float4*)&h[(size_t)(m0 + row) * HID + col];
        As[row * PADH + col + 0] = v.x;
        As[row * PADH + col + 1] = v.y;
        As[row * PADH + col + 2] = v.z;
        As[row * PADH + col + 3] = v.w;
    }
    __syncthreads();

    const int wave = tid >> 5, lane = tid & 31;
    const int half = lane >> 4, li = lane & 15;
    const int n0 = wave * 16;                     // 2 waves cover N=32

    v8f acc = {};
    for (int k0 = 0; k0 < HID; k0 += 4) {
        const int kk = k0 + half * 2;
        v2f a; a.x = As[li * PADH + kk]; a.y = As[li * PADH + kk + 1];
        const float* bp = &w_out[kk * NCLS + n0 + li];
        v2f b; b.x = bp[0]; b.y = bp[NCLS];
        acc = wmma_f32(a, b, acc);
    }
#pragma unroll
    for (int v = 0; v < 8; ++v) {
        const int row = v + half * 8;
        out[(size_t)(m0 + row) * NCLS + n0 + li] = acc[v];
    }
}

extern "C" void kernel_launch(void* const* d_in, const int* in_sizes, int n_in,
                              void* d_out, int out_size, void* d_ws, size_t ws_size,
                              hipStream_t stream) {
    const float* x     = (const float*)d_in[0];
    const int*   ei    = (const int*)d_in[1];
    const float* w_in  = (const float*)d_in[2];
    const float* w_out = (const float*)d_in[3];
    const float* convw = (const float*)d_in[4];
    const float* w_ih  = (const float*)d_in[5];
    const float* w_hh  = (const float*)d_in[6];
    const float* b_ih  = (const float*)d_in[7];
    const float* b_hh  = (const float*)d_in[8];
    float* out = (float*)d_out;

    float* h   = (float*)d_ws;                    // 50000*64
    float* m   = h + (size_t)N_NODES * HID;       // 50000*64
    float* agg = m + (size_t)N_NODES * HID;       // 50000*64   (38.4 MB total)

    const dim3 g(N_NODES / 16);                   // 3125 row tiles (exact)

    k_input_gemm<<<g, 128, 0, stream>>>(x, w_in, h);
    for (int l = 0; l < LAYERS; ++l) {
        k_transform<<<g, 128, 0, stream>>>(h, convw + l * HID * HID, m, agg);
        k_scatter<<<(N_EDGES * 16) / 256, 256, 0, stream>>>(ei, m, agg);
        k_gru<<<g, 384, 0, stream>>>(h, agg, w_ih, w_hh, b_ih, b_hh);
    }
    k_output<<<g, 64, 0, stream>>>(h, w_out, out);
}